// ModelStep_66013647339689
// MI455X (gfx1250) — compile-verified
//
#include <hip/hip_runtime.h>

// ---------------------------------------------------------------------------
// Batched Dopri5 (jax.experimental.ode.odeint-faithful) for 8192 independent
// 8-species Michaelis-Menten trajectories.  One thread = one trajectory.
// Input staged global->LDS through the CDNA5 Tensor Data Mover (TENSORcnt).
// ---------------------------------------------------------------------------

typedef unsigned int v4u __attribute__((ext_vector_type(4)));
typedef int          v8i __attribute__((ext_vector_type(8)));
typedef int          v4i __attribute__((ext_vector_type(4)));

#ifndef __has_builtin
#define __has_builtin(x) 0
#endif

#if __has_builtin(__builtin_amdgcn_tensor_load_to_lds) && \
    __has_builtin(__builtin_amdgcn_s_wait_tensorcnt)
#define USE_TDM 1
#else
#define USE_TDM 0
#endif

#define BLOCK 64   // 2 wave32 waves per workgroup; 128 blocks over the WGPs

// dy/dt for one trajectory; STOICH@v folded analytically (rows are {+1,-1}).
__device__ __forceinline__ void rate_fn(const float* __restrict__ c,
                                        const float* __restrict__ y,
                                        float* __restrict__ dy) {
  float v1 = c[1] * y[0] / (c[2] + y[0]);
  float v2 = c[3] * y[1] / (c[4] + y[1]);
  float v3 = c[5] * y[0] * y[2] / (c[6] + y[2]);
  float v4 = c[7] * y[3] / (c[8] + y[3]);
  float v5 = c[9] * y[3] * y[4] / (c[10] + y[4]);
  float v6 = c[11] * y[5] / (c[12] + y[5]);
  float v7 = c[13] * y[5] * y[6] / (c[14] + y[6]);
  float v8 = c[15] * y[7] / c[16] / (1.0f + y[7] / c[16] + c[17] / c[18]);
  float d01 = v2 - v1, d23 = v4 - v3, d45 = v6 - v5, d67 = v8 - v7;
  dy[0] = d01; dy[1] = -d01;
  dy[2] = d23; dy[3] = -d23;
  dy[4] = d45; dy[5] = -d45;
  dy[6] = d67; dy[7] = -d67;
}

__global__ __launch_bounds__(BLOCK) void ode_dopri5_kernel(
    const float* __restrict__ y_in, const float* __restrict__ cg,
    const float* __restrict__ t_in, const float* __restrict__ dt_in,
    float* __restrict__ y_out, int B) {
  __shared__ float lds_y[BLOCK * 8];
  const int nElem = BLOCK * 8;
  const float* gsrc = y_in + (size_t)blockIdx.x * (size_t)nElem;

#if USE_TDM
  // Wave 0 issues one TDM descriptor pulling this block's 2KB tile into LDS.
  // D# group0: count=1, lds_addr, global_addr(57b), type=2.
  // D# group1: data_size=4B, tensor_dim0=tile_dim0=nElem (1-row 2D tile),
  //            tensor_dim0_stride=nElem, tensor_dim1=1, groups 2/3 zeroed.
  if (threadIdx.x < 32) {
    unsigned long long ga = (unsigned long long)(const void*)gsrc;
    // Flat shared addresses carry the LDS byte offset in their low 32 bits.
    unsigned lds_off = (unsigned)(unsigned long long)(void*)&lds_y[0];
    v4u g0;
    g0.x = 1u;                                  // count=1 (valid descriptor)
    g0.y = lds_off;                             // LDS byte address
    g0.z = (unsigned)ga;                        // global_addr[31:0]
    g0.w = (unsigned)(ga >> 32) | (2u << 30);   // global_addr[56:32] | type=2
    unsigned n = (unsigned)nElem;
    v8i g1;
    g1[0] = (int)(2u << 16);                    // data_size = 4 bytes
    g1[1] = (int)((n & 0xFFFFu) << 16);         // tensor_dim0[15:0]
    g1[2] = (int)(((n >> 16) & 0xFFFFu) | (1u << 16)); // dim0[31:16] | tensor_dim1=1
    g1[3] = (int)((n & 0xFFFFu) << 16);         // tile_dim0 = nElem
    g1[4] = 1;                                  // tile_dim1 = 1, tile_dim2 = 0
    g1[5] = (int)n;                             // tensor_dim0_stride[31:0]
    g1[6] = 0;
    g1[7] = 0;
    v4i gz = {0, 0, 0, 0};
#if defined(__clang_major__) && (__clang_major__ >= 23)
    v8i gz8 = {0, 0, 0, 0, 0, 0, 0, 0};
    __builtin_amdgcn_tensor_load_to_lds(g0, g1, gz, gz, gz8, 0);
#else
    __builtin_amdgcn_tensor_load_to_lds(g0, g1, gz, gz, 0);
#endif
    __builtin_amdgcn_s_wait_tensorcnt(0);       // data landed before we signal
  }
  __syncthreads();
#else
  for (int idx = threadIdx.x; idx < nElem; idx += BLOCK) lds_y[idx] = gsrc[idx];
  __syncthreads();
#endif

  const int gtid = blockIdx.x * BLOCK + threadIdx.x;
  if (gtid >= B) return;

  float y[8];
#pragma unroll
  for (int i = 0; i < 8; ++i) y[i] = lds_y[threadIdx.x * 8 + i];

  float c[19];
#pragma unroll
  for (int i = 0; i < 19; ++i) c[i] = cg[i];   // uniform -> s_loads

  const float t0 = t_in[0];
  const float target = t0 + dt_in[0];
  const float atol = 1e-6f, rtol = 1e-12f;

  // ---- jax initial_step_size (order=4) ----
  float f[8];
  rate_fn(c, y, f);
  float scale[8], d0 = 0.f, d1 = 0.f;
#pragma unroll
  for (int i = 0; i < 8; ++i) {
    scale[i] = atol + fabsf(y[i]) * rtol;
    float a = y[i] / scale[i], b = f[i] / scale[i];
    d0 += a * a; d1 += b * b;
  }
  d0 = sqrtf(d0); d1 = sqrtf(d1);
  float h0 = ((d0 < 1e-5f) || (d1 < 1e-5f)) ? 1e-6f : 0.01f * d0 / d1;
  float ytmp[8], ftmp[8];
#pragma unroll
  for (int i = 0; i < 8; ++i) ytmp[i] = y[i] + h0 * f[i];
  rate_fn(c, ytmp, ftmp);
  float d2 = 0.f;
#pragma unroll
  for (int i = 0; i < 8; ++i) { float r = (ftmp[i] - f[i]) / scale[i]; d2 += r * r; }
  d2 = sqrtf(d2) / h0;
  float dt = ((d1 <= 1e-15f) && (d2 <= 1e-15f))
                 ? fmaxf(1e-6f, h0 * 1e-3f)
                 : powf(0.01f / fmaxf(d1, d2), 0.2f);
  dt = fminf(100.f * h0, dt);

  // ---- Dopri5 tableau (as in jax.experimental.ode) ----
  const float BETA[6][6] = {
      {0.2f, 0.f, 0.f, 0.f, 0.f, 0.f},
      {3.f / 40.f, 9.f / 40.f, 0.f, 0.f, 0.f, 0.f},
      {44.f / 45.f, -56.f / 15.f, 32.f / 9.f, 0.f, 0.f, 0.f},
      {19372.f / 6561.f, -25360.f / 2187.f, 64448.f / 6561.f, -212.f / 729.f, 0.f, 0.f},
      {9017.f / 3168.f, -355.f / 33.f, 46732.f / 5247.f, 49.f / 176.f, -5103.f / 18656.f, 0.f},
      {35.f / 384.f, 0.f, 500.f / 1113.f, 125.f / 192.f, -2187.f / 6784.f, 11.f / 84.f}};
  const float CERR[7] = {
      (float)(35.0 / 384.0 - 1951.0 / 21600.0), 0.f,
      (float)(500.0 / 1113.0 - 22642.0 / 50085.0),
      (float)(125.0 / 192.0 - 451.0 / 720.0),
      (float)(-2187.0 / 6784.0 + 12231.0 / 42400.0),
      (float)(11.0 / 84.0 - 649.0 / 6300.0),
      (float)(-1.0 / 60.0)};
  const float CMID[7] = {
      (float)(6025192743.0 / 30085553152.0 / 2.0), 0.f,
      (float)(51252292925.0 / 65400821598.0 / 2.0),
      (float)(-2691868925.0 / 45128329728.0 / 2.0),
      (float)(187940372067.0 / 1594534317056.0 / 2.0),
      (float)(-1776094331.0 / 19743644256.0 / 2.0),
      (float)(11237099.0 / 235043384.0 / 2.0)};

  float t = t0, last_t = t0;
  float yprev[8], fprev[8], ymid[8];
#pragma unroll
  for (int i = 0; i < 8; ++i) { yprev[i] = y[i]; fprev[i] = f[i]; ymid[i] = y[i]; }
  bool have = false;

  int it = 0;
  while ((t < target) && (dt > 0.f) && (it < 5000000)) {
    ++it;
    float k[7][8];
#pragma unroll
    for (int i = 0; i < 8; ++i) k[0][i] = f[i];
    float ys[8];
#pragma unroll
    for (int s = 1; s < 7; ++s) {
#pragma unroll
      for (int i = 0; i < 8; ++i) {
        float acc = 0.f;
#pragma unroll
        for (int j = 0; j < 6; ++j)
          if (j < s) acc += BETA[s - 1][j] * k[j][i];
        ys[i] = y[i] + dt * acc;
      }
      rate_fn(c, ys, k[s]);
    }
    // FSAL: stage-7 input IS y_next (c_sol == BETA row 6, c_sol[6]==0).
    float ynext[8], yerr[8];
#pragma unroll
    for (int i = 0; i < 8; ++i) {
      ynext[i] = ys[i];
      float acc = 0.f;
#pragma unroll
      for (int j = 0; j < 7; ++j) acc += CERR[j] * k[j][i];
      yerr[i] = dt * acc;
    }
    // mean_error_ratio = mean((err/tol)^2); accept iff <= 1
    float r2 = 0.f;
#pragma unroll
    for (int i = 0; i < 8; ++i) {
      float tol = atol + rtol * fmaxf(fabsf(y[i]), fabsf(ynext[i]));
      float r = yerr[i] / tol;
      r2 += r * r;
    }
    r2 *= 0.125f;
    const bool accept = (r2 <= 1.f);
    // optimal_step_size: safety=0.9, ifactor=10, dfactor=0.2, order=5
    float dfac = (r2 < 1.f) ? 1.f : 0.2f;
    float factor = fmaxf(0.1f, fminf(powf(sqrtf(r2), 0.2f) / 0.9f, 1.f / dfac));
    float dtnew = (r2 == 0.f) ? dt * 10.f : dt / factor;

    if (accept) {
#pragma unroll
      for (int i = 0; i < 8; ++i) {
        float acc = 0.f;
#pragma unroll
        for (int j = 0; j < 7; ++j) acc += CMID[j] * k[j][i];
        ymid[i] = y[i] + dt * acc;   // dps_c_mid midpoint for interp fit
        yprev[i] = y[i];
        fprev[i] = f[i];
        y[i] = ynext[i];
        f[i] = k[6][i];
      }
      last_t = t;
      t = t + dt;
      have = true;
    }
    dt = fmaxf(dtnew, 0.f);
  }

  // ---- 4th-order interpolation back to target (jax overshoots, then fits) ----
  float yout[8];
  if (have && (t > last_t)) {
    float h = t - last_t;
    float x = (target - last_t) / h;
#pragma unroll
    for (int i = 0; i < 8; ++i) {
      float a = -2.f * h * fprev[i] + 2.f * h * f[i] - 8.f * yprev[i] - 8.f * y[i] + 16.f * ymid[i];
      float b =  5.f * h * fprev[i] - 3.f * h * f[i] + 18.f * yprev[i] + 14.f * y[i] - 32.f * ymid[i];
      float cc = -4.f * h * fprev[i] +       h * f[i] - 11.f * yprev[i] - 5.f * y[i] + 16.f * ymid[i];
      float d =        h * fprev[i];
      float e = yprev[i];
      yout[i] = (((a * x + b) * x + cc) * x + d) * x + e;
    }
  } else {
#pragma unroll
    for (int i = 0; i < 8; ++i) yout[i] = y[i];
  }

  float4* outv = (float4*)(y_out + (size_t)gtid * 8);
  outv[0] = make_float4(yout[0], yout[1], yout[2], yout[3]);
  outv[1] = make_float4(yout[4], yout[5], yout[6], yout[7]);
}

// Writes the trailing tuple members: c (19 floats) and t_new (1 float).
__global__ void tail_kernel(const float* __restrict__ cg,
                            const float* __restrict__ t_in,
                            const float* __restrict__ dt_in,
                            float* __restrict__ out_tail, int nc) {
  int i = threadIdx.x;
  if (i < nc) out_tail[i] = cg[i];
  if (i == 0) out_tail[nc] = t_in[0] + dt_in[0];
}

extern "C" void kernel_launch(void* const* d_in, const int* in_sizes, int n_in,
                              void* d_out, int out_size, void* d_ws, size_t ws_size,
                              hipStream_t stream) {
  (void)n_in; (void)out_size; (void)d_ws; (void)ws_size;
  const float* y  = (const float*)d_in[0];   // (B, 8) float32
  // d_in[1] = w, 0 elements
  const float* c  = (const float*)d_in[2];   // (19,)
  const float* t  = (const float*)d_in[3];   // scalar
  const float* dT = (const float*)d_in[4];   // scalar
  float* out = (float*)d_out;

  const int B = in_sizes[0] / 8;             // 8192 trajectories
  const int nc = in_sizes[2];                // 19
  dim3 block(BLOCK), grid((B + BLOCK - 1) / BLOCK);
  ode_dopri5_kernel<<<grid, block, 0, stream>>>(y, c, t, dT, out, B);

  // out layout: y_new (B*8) | w (in_sizes[1]==0) | c (nc) | t_new (1)
  float* out_tail = out + (size_t)B * 8 + (size_t)in_sizes[1];
  tail_kernel<<<1, 32, 0, stream>>>(c, t, dT, out_tail, nc);
}